// CombinedDynamicMarginLoss_4526895529993
// MI455X (gfx1250) — compile-verified
//
#include <hip/hip_runtime.h>
#include <stdint.h>

typedef float v4f __attribute__((ext_vector_type(4)));

#define TPB     256
#define CHUNK   4                      // float4 loads per thread per tile
#define TILE_V4 (TPB * CHUNK)          // 1024 float4 = 4096 floats = 16 KB per tile

#define K_S       64.0f
#define K_ALPHA   0.1f
#define K_M2      0.5f
#define K_THRESH  0.4f
#define K_NEGBIG  (-1.0e9f)

// ---- CDNA5 async global->LDS copy (ASYNCcnt path), per-lane b128 ----
__device__ __forceinline__ void async_ld_b128(const void* g, void* lds) {
  // VDST = per-lane LDS byte offset (low 32 bits of the flat shared address),
  // VADDR = per-lane 64-bit global address, GV mode (saddr = off).
  unsigned       lofs = (unsigned)(unsigned long long)(uintptr_t)lds;
  unsigned long long ga = (unsigned long long)(uintptr_t)g;
  asm volatile("global_load_async_to_lds_b128 %0, %1, off"
               :: "v"(lofs), "v"(ga)
               : "memory");
}

template <int N>
__device__ __forceinline__ void wait_async() {
#if __has_builtin(__builtin_amdgcn_s_wait_asynccnt)
  __builtin_amdgcn_s_wait_asynccnt(N);
#else
  asm volatile("s_wait_asynccnt %0" :: "i"(N) : "memory");
#endif
  asm volatile("" ::: "memory");   // keep LDS reads after the wait
}

__device__ __forceinline__ float filt(float x) {
  return (x > K_THRESH) ? 0.0f : x;
}

__device__ __forceinline__ void proc1(float x, int j, int pos,
                                      float* __restrict__ op, float& lmax) {
  op[j] = x * K_S;
  lmax  = fmaxf(lmax, (j == pos) ? K_NEGBIG : filt(x));
}

__global__ __launch_bounds__(TPB)
void margin_scan_kernel(const float* __restrict__ logits,
                        const int*   __restrict__ labels,
                        float* __restrict__ out,
                        float* __restrict__ phiS,
                        int C) {
  const int    row  = blockIdx.x;
  const int    tid  = threadIdx.x;
  const size_t base = (size_t)row * (size_t)C;
  const float* rp   = logits + base;
  float*       op   = out + base;
  const int    pos  = labels[row];

  __shared__ v4f   buf[2][TILE_V4];   // 32 KB double buffer
  __shared__ float red[TPB];

  // Align the vectorized region to 16B in the flat array.
  const int j0    = (int)((4u - (unsigned)(base & 3ull)) & 3u);
  const int nv4   = (C - j0) >> 2;
  const int nfull = nv4 / TILE_V4;
  const v4f* gv     = (const v4f*)(rp + j0);
  v4f*       govout = (v4f*)(op + j0);

  float lmax = K_NEGBIG;

  // prologue scalars (<=3)
  if (tid < j0) proc1(rp[tid], tid, pos, op, lmax);

  // ---- async double-buffered tile pipeline ----
  if (nfull > 0) {
    for (int u = 0; u < CHUNK; ++u) {
      const int slot = u * TPB + tid;
      async_ld_b128(gv + slot, &buf[0][slot]);
    }
    for (int t = 0; t < nfull; ++t) {
      const int b = t & 1;
      if (t + 1 < nfull) {
        const int nb = (t + 1) & 1;
        const int vb = (t + 1) * TILE_V4;
        for (int u = 0; u < CHUNK; ++u) {
          const int slot = u * TPB + tid;
          async_ld_b128(gv + vb + slot, &buf[nb][slot]);
        }
        wait_async<CHUNK>();   // older tile's CHUNK loads retired (in-order)
      } else {
        wait_async<0>();
      }
      const int vb = t * TILE_V4;
      for (int u = 0; u < CHUNK; ++u) {
        const int slot = u * TPB + tid;
        v4f x = buf[b][slot];
        v4f y = x * K_S;
        __builtin_nontemporal_store(y, govout + vb + slot);
        const int e = j0 + ((vb + slot) << 2);
        lmax = fmaxf(lmax, (e + 0 == pos) ? K_NEGBIG : filt(x.x));
        lmax = fmaxf(lmax, (e + 1 == pos) ? K_NEGBIG : filt(x.y));
        lmax = fmaxf(lmax, (e + 2 == pos) ? K_NEGBIG : filt(x.z));
        lmax = fmaxf(lmax, (e + 3 == pos) ? K_NEGBIG : filt(x.w));
      }
    }
  }

  // remainder float4s (last partial tile) via direct NT loads
  for (int v = nfull * TILE_V4 + tid; v < nv4; v += TPB) {
    v4f x = __builtin_nontemporal_load(gv + v);
    v4f y = x * K_S;
    __builtin_nontemporal_store(y, govout + v);
    const int e = j0 + (v << 2);
    lmax = fmaxf(lmax, (e + 0 == pos) ? K_NEGBIG : filt(x.x));
    lmax = fmaxf(lmax, (e + 1 == pos) ? K_NEGBIG : filt(x.y));
    lmax = fmaxf(lmax, (e + 2 == pos) ? K_NEGBIG : filt(x.z));
    lmax = fmaxf(lmax, (e + 3 == pos) ? K_NEGBIG : filt(x.w));
  }

  // tail scalars (<=3)
  const int jt = j0 + (nv4 << 2);
  if (jt + tid < C) proc1(rp[jt + tid], jt + tid, pos, op, lmax);

  // ---- block max reduction ----
  red[tid] = lmax;
  __syncthreads();
  for (int s = TPB / 2; s > 0; s >>= 1) {
    if (tid < s) red[tid] = fmaxf(red[tid], red[tid + s]);
    __syncthreads();
  }

  if (tid == 0) {
    float val = 0.0f;
    if (pos != -1) {
      const float mo  = red[0];                 // max_other (filtered, pos masked)
      const float cy  = rp[pos];                // cos_y (filter exempts pos column)
      const float h   = 1.0f - (cy - mo);
      const float mi  = K_M2 + K_ALPHA * h;     // M2 + alpha*h
      const float th  = acosf(fminf(fmaxf(cy, -1.0f), 1.0f));
      const float ph  = cosf(th + mi);          // M1=1, M3=0
      const float fin = (ph < cy) ? ph : cy;
      val = fin * K_S;
    }
    phiS[row] = val;
  }
}

// Scatter the margined value into the label column (runs after kernel 1,
// so stream ordering guarantees visibility of the streaming stores).
__global__ void margin_fix_kernel(const float* __restrict__ phiS,
                                  const int*   __restrict__ labels,
                                  float* __restrict__ out,
                                  int N, int C) {
  const int i = blockIdx.x * blockDim.x + threadIdx.x;
  if (i < N) {
    const int lab = labels[i];
    if (lab != -1) out[(size_t)i * (size_t)C + (size_t)lab] = phiS[i];
  }
}

extern "C" void kernel_launch(void* const* d_in, const int* in_sizes, int n_in,
                              void* d_out, int out_size, void* d_ws, size_t ws_size,
                              hipStream_t stream) {
  const float* logits = (const float*)d_in[0];
  const int*   labels = (const int*)d_in[1];
  float*       out    = (float*)d_out;
  float*       phiS   = (float*)d_ws;

  const int N = in_sizes[1];
  const int C = in_sizes[0] / N;

  margin_scan_kernel<<<N, TPB, 0, stream>>>(logits, labels, out, phiS, C);
  margin_fix_kernel<<<(N + TPB - 1) / TPB, TPB, 0, stream>>>(phiS, labels, out, N, C);
}